// GCNnet_22720376996600
// MI455X (gfx1250) — compile-verified
//
#include <hip/hip_runtime.h>
#include <hip/hip_bf16.h>

// ---------------------------------------------------------------------------
// GCN forward for MI455X (gfx1250, wave32).
//   * edge message passing = fp32 L2 atomics (output arrays fit in 192MB L2)
//   * GEMMs = V_WMMA_F32_16X16X4_F32 (exact fp32, 16x16 tile per wave)
//   * layer-1 aggregates x (32-wide) BEFORE the GEMM (Agg is linear) -> 8x
//     less edge traffic; b1/b2 fold out under train-mode BatchNorm.
//   * global max pool via integer atomicMax on non-negative float bits.
// ---------------------------------------------------------------------------

#define N_NODES   100000
#define N_EDGES   1600000
#define NUM_GRAPHS 64
#define BN_EPS    1e-5f

typedef __attribute__((ext_vector_type(2))) float v2f;
typedef __attribute__((ext_vector_type(8))) float v8f;

__device__ __forceinline__ void atomAddF(float* p, float v) {
  unsafeAtomicAdd(p, v);   // -> global_atomic_add_f32
}

// ---------------- degree / normalization ------------------------------------
__global__ void k_deg_init(float* deg) {            // self-loop weight = 1
  int n = blockIdx.x * 256 + threadIdx.x;
  if (n < N_NODES) deg[n] = 1.0f;
}

__global__ void k_deg_accum(const long long* __restrict__ col,
                            const float* __restrict__ ew, float* deg) {
  int e = blockIdx.x * 256 + threadIdx.x;
  if (e < N_EDGES) atomAddF(&deg[col[e]], ew[e]);
}

__global__ void k_dinv(float* deg) {                // in place: deg -> D^{-1/2}
  int n = blockIdx.x * 256 + threadIdx.x;
  if (n < N_NODES) {
    float d = deg[n];
    deg[n] = d > 0.0f ? rsqrtf(d) : 0.0f;
  }
}

__global__ void k_norm(const long long* __restrict__ row,
                       const long long* __restrict__ col,
                       const float* __restrict__ ew,
                       const float* __restrict__ dinv,
                       float* __restrict__ norm) {
  int e = blockIdx.x * 256 + threadIdx.x;
  if (e < N_EDGES) norm[e] = dinv[row[e]] * ew[e] * dinv[col[e]];
}

// ---------------- aggregation: out[c] = dinv[c]^2*h[c] (+b) + sum norm*h[row]
template <int F, bool BIAS>
__global__ void k_selfinit(const float* __restrict__ h,
                           const float* __restrict__ dinv,
                           const float* __restrict__ bias,
                           float* __restrict__ out) {
  constexpr int Q = F / 4;
  int i = blockIdx.x * 256 + threadIdx.x;           // over N*F/4 float4s
  if (i >= N_NODES * Q) return;
  int n = i / Q, c4 = (i % Q) * 4;
  float d = dinv[n]; d *= d;                        // self-loop norm = dinv^2
  float4 v = ((const float4*)h)[i];
  v.x *= d; v.y *= d; v.z *= d; v.w *= d;
  if (BIAS) { v.x += bias[c4]; v.y += bias[c4+1]; v.z += bias[c4+2]; v.w += bias[c4+3]; }
  ((float4*)out)[i] = v;
}

template <int F>
__global__ void __launch_bounds__(256)
k_scatter(const long long* __restrict__ row, const long long* __restrict__ col,
          const float* __restrict__ norm, const float* __restrict__ h,
          float* __restrict__ out) {
  constexpr int TPE = F / 4;                        // threads per edge
  int idx = blockIdx.x * 256 + threadIdx.x;
  int e = idx / TPE, t = idx % TPE;
  if (e >= N_EDGES) return;
  long long r = row[e], c = col[e];
  float w = norm[e];
  float4 v = ((const float4*)(h + (size_t)r * F))[t];
  float* o = out + (size_t)c * F + t * 4;           // hot set lives in L2
  atomAddF(o + 0, w * v.x);
  atomAddF(o + 1, w * v.y);
  atomAddF(o + 2, w * v.z);
  atomAddF(o + 3, w * v.w);
}

// ---------------- fp32 WMMA GEMM: C[M,N] = A[M,K] @ B[K,N] ------------------
// One wave -> one 16x16 tile via V_WMMA_F32_16X16X4_F32, K-step 4.
// A frag (16x4): lane L holds M=L&15; VGPR0 K=k+2h, VGPR1 K=k+1+2h (h=L>>4).
// B frag (4x16): lane L holds N=L&15; VGPR v holds K=k+v+2h.
// C/D (16x16):   lane L holds N=L&15; VGPR v holds M=v+8h.
__global__ void __launch_bounds__(256)
k_gemm_wmma(const float* __restrict__ A, const float* __restrict__ B,
            float* __restrict__ C, int M, int K, int Nout) {
  const int lane = threadIdx.x & 31;
  const int wave = threadIdx.x >> 5;                // 8 waves -> 128 cols/block
  const int col0 = blockIdx.x * 128 + wave * 16;
  const int row0 = blockIdx.y * 16;
  if (col0 >= Nout || row0 >= M) return;            // uniform per wave
  const int h = lane >> 4;
  const int l = lane & 15;
  v8f acc = {};
  const float* Arow = A + (size_t)(row0 + l) * K;
  const float* Bcol = B + col0 + l;
  for (int k = 0; k < K; k += 4) {
    v2f a, b;
    a.x = Arow[k + 2 * h + 0];
    a.y = Arow[k + 2 * h + 1];
    b.x = Bcol[(size_t)(k + 2 * h + 0) * Nout];
    b.y = Bcol[(size_t)(k + 2 * h + 1) * Nout];
    acc = __builtin_amdgcn_wmma_f32_16x16x4_f32(false, a, false, b,
                                                (short)0, acc, false, false);
  }
  float* Crow = C + (size_t)row0 * Nout + col0 + l;
#pragma unroll
  for (int v = 0; v < 8; ++v) Crow[(size_t)(v + 8 * h) * Nout] = acc[v];
}

// ---------------- BatchNorm (train-mode stats) + ReLU -----------------------
__global__ void k_zero(float* p, int n) {
  int i = blockIdx.x * 256 + threadIdx.x;
  if (i < n) p[i] = 0.0f;
}

__global__ void k_bn_stats(const float* __restrict__ h, float* sum, float* sq) {
  int c = threadIdx.x;                               // 256 channels
  int r0 = blockIdx.x * 512;
  int r1 = min(r0 + 512, N_NODES);
  float s = 0.f, q = 0.f;
  for (int r = r0; r < r1; ++r) {
    float v = h[(size_t)r * 256 + c];
    s += v; q += v * v;
  }
  atomAddF(&sum[c], s);
  atomAddF(&sq[c], q);
}

__global__ void k_bn_coef(const float* sum, const float* sq, const float* g,
                          const float* be, float* scale, float* shift) {
  int c = threadIdx.x;
  float inv = 1.0f / (float)N_NODES;
  float mu = sum[c] * inv;
  float var = sq[c] * inv - mu * mu;                 // biased var
  float sc = g[c] * rsqrtf(var + BN_EPS);
  scale[c] = sc;
  shift[c] = be[c] - mu * sc;
}

__global__ void k_bn_apply_relu(float* __restrict__ x,
                                const float* __restrict__ scale,
                                const float* __restrict__ shift) {
  int i = blockIdx.x * 256 + threadIdx.x;            // over N*256/4 float4s
  if (i >= N_NODES * 64) return;
  int c4 = (i & 63) * 4;
  float4 v = ((float4*)x)[i];
  v.x = fmaxf(v.x * scale[c4 + 0] + shift[c4 + 0], 0.f);
  v.y = fmaxf(v.y * scale[c4 + 1] + shift[c4 + 1], 0.f);
  v.z = fmaxf(v.z * scale[c4 + 2] + shift[c4 + 2], 0.f);
  v.w = fmaxf(v.w * scale[c4 + 3] + shift[c4 + 3], 0.f);
  ((float4*)x)[i] = v;
}

// ---------------- global max pool (per-graph, 128 ch) -----------------------
// relu first -> values >= 0 -> IEEE order == int order -> integer atomicMax.
__global__ void k_pool(const float* __restrict__ a,
                       const long long* __restrict__ batch, int* pool) {
  int i = blockIdx.x * 256 + threadIdx.x;            // over N*128
  if (i >= N_NODES * 128) return;
  int n = i >> 7, c = i & 127;
  float v = fmaxf(a[i], 0.0f);
  int g = (int)batch[n];
  atomicMax(&pool[g * 128 + c], __float_as_int(v));
}

// ---------------- MLP head: [64,128] -> relu(@W1+b1) -> @W2+b2 -> [64,2] ----
__global__ void k_fc(const float* __restrict__ pool,
                     const float* __restrict__ fcW1, const float* __restrict__ fcb1,
                     const float* __restrict__ fcW2, const float* __restrict__ fcb2,
                     float* __restrict__ out) {
  __shared__ float sp[128];
  __shared__ float sh[64];
  int g = blockIdx.x, t = threadIdx.x;               // 64 threads
  sp[t] = pool[g * 128 + t];
  sp[t + 64] = pool[g * 128 + 64 + t];
  __syncthreads();
  float acc = fcb1[t];
  for (int k = 0; k < 128; ++k) acc += sp[k] * fcW1[k * 64 + t];
  sh[t] = fmaxf(acc, 0.0f);
  __syncthreads();
  if (t < 2) {
    float o = fcb2[t];
    for (int j = 0; j < 64; ++j) o += sh[j] * fcW2[j * 2 + t];
    out[g * 2 + t] = o;
  }
}

// ---------------------------------------------------------------------------
extern "C" void kernel_launch(void* const* d_in, const int* in_sizes, int n_in,
                              void* d_out, int out_size, void* d_ws, size_t ws_size,
                              hipStream_t stream) {
  const float*     x    = (const float*)d_in[0];       // [N,32]
  const long long* ei   = (const long long*)d_in[1];   // [2,E]
  const float*     ew   = (const float*)d_in[2];       // [E]
  const long long* batch= (const long long*)d_in[3];   // [N]
  const float* W1  = (const float*)d_in[4];            // [32,256]   (b1 folds out)
  const float* W2  = (const float*)d_in[6];            // [256,256]  (b2 folds out)
  const float* W3  = (const float*)d_in[8];            // [256,128]
  const float* b3  = (const float*)d_in[9];            // [128]
  const float* g1  = (const float*)d_in[10];
  const float* be1 = (const float*)d_in[11];
  const float* g2  = (const float*)d_in[12];
  const float* be2 = (const float*)d_in[13];
  const float* fcW1 = (const float*)d_in[14];
  const float* fcb1 = (const float*)d_in[15];
  const float* fcW2 = (const float*)d_in[16];
  const float* fcb2 = (const float*)d_in[17];
  float* out = (float*)d_out;                          // [64,2]

  const long long* row = ei;
  const long long* col = ei + N_EDGES;

  // workspace layout (~212 MB)
  float* ws    = (float*)d_ws;
  float* dinv  = ws;                                   // N
  float* norm  = dinv + N_NODES;                       // E
  float* bufA  = norm + N_EDGES;                       // N*256
  float* bufB  = bufA + (size_t)N_NODES * 256;         // N*256
  float* aggx  = bufB;                                 // N*32 (aliases bufB)
  float* stat  = bufB + (size_t)N_NODES * 256;         // sum|sq|scale|shift
  float* s_sum = stat, *s_sq = stat + 256, *s_scale = stat + 512, *s_shift = stat + 768;
  float* pool  = stat + 1024;                          // 64*128

  const int nb_nodes = (N_NODES + 255) / 256;
  const int nb_edges = N_EDGES / 256;

  // ---- gcn_norm ----
  k_deg_init <<<nb_nodes, 256, 0, stream>>>(dinv);
  k_deg_accum<<<nb_edges, 256, 0, stream>>>(col, ew, dinv);
  k_dinv     <<<nb_nodes, 256, 0, stream>>>(dinv);
  k_norm     <<<nb_edges, 256, 0, stream>>>(row, col, ew, dinv, norm);

  // ---- layer 1: Agg(x) (32-wide!) then GEMM, BN+ReLU ----
  k_selfinit<32, false><<<(N_NODES * 8 + 255) / 256, 256, 0, stream>>>(x, dinv, nullptr, aggx);
  k_scatter<32><<<N_EDGES * 8 / 256, 256, 0, stream>>>(row, col, norm, x, aggx);
  k_gemm_wmma<<<dim3(2, N_NODES / 16), 256, 0, stream>>>(aggx, W1, bufA, N_NODES, 32, 256);
  k_zero<<<2, 256, 0, stream>>>(s_sum, 512);
  k_bn_stats<<<(N_NODES + 511) / 512, 256, 0, stream>>>(bufA, s_sum, s_sq);
  k_bn_coef<<<1, 256, 0, stream>>>(s_sum, s_sq, g1, be1, s_scale, s_shift);
  k_bn_apply_relu<<<(N_NODES * 64 + 255) / 256, 256, 0, stream>>>(bufA, s_scale, s_shift);

  // ---- layer 2: GEMM (256x256) then Agg, BN+ReLU ----
  k_gemm_wmma<<<dim3(2, N_NODES / 16), 256, 0, stream>>>(bufA, W2, bufB, N_NODES, 256, 256);
  k_selfinit<256, false><<<(N_NODES * 64 + 255) / 256, 256, 0, stream>>>(bufB, dinv, nullptr, bufA);
  k_scatter<256><<<N_EDGES * 64 / 256, 256, 0, stream>>>(row, col, norm, bufB, bufA);
  k_zero<<<2, 256, 0, stream>>>(s_sum, 512);
  k_bn_stats<<<(N_NODES + 511) / 512, 256, 0, stream>>>(bufA, s_sum, s_sq);
  k_bn_coef<<<1, 256, 0, stream>>>(s_sum, s_sq, g2, be2, s_scale, s_shift);
  k_bn_apply_relu<<<(N_NODES * 64 + 255) / 256, 256, 0, stream>>>(bufA, s_scale, s_shift);

  // ---- layer 3: GEMM (256x128) then Agg (+b3); ReLU fused into pool ----
  float* h3   = bufB;                                  // N*128
  float* agg3 = bufB + (size_t)N_NODES * 128;          // N*128
  k_gemm_wmma<<<dim3(1, N_NODES / 16), 256, 0, stream>>>(bufA, W3, h3, N_NODES, 256, 128);
  k_selfinit<128, true><<<(N_NODES * 32 + 255) / 256, 256, 0, stream>>>(h3, dinv, b3, agg3);
  k_scatter<128><<<N_EDGES * 32 / 256, 256, 0, stream>>>(row, col, norm, h3, agg3);

  // ---- pool + head ----
  k_zero<<<32, 256, 0, stream>>>(pool, NUM_GRAPHS * 128);
  k_pool<<<(N_NODES * 128 + 255) / 256, 256, 0, stream>>>(agg3, batch, (int*)pool);
  k_fc<<<NUM_GRAPHS, 64, 0, stream>>>(pool, fcW1, fcb1, fcW2, fcb2, out);
}